// Unfold_2757369004156
// MI455X (gfx1250) — compile-verified
//
#include <hip/hip_runtime.h>

#define HH   48
#define WD   (48 * 48)            // 2304 floats per h-row (w*d plane row)
#define HWD  (48 * 48 * 48)       // 110592 floats per channel
#define TH   2                    // output h-rows produced per block
#define ROWS (TH + 2)             // staged rows incl. +/-1 halo
#define CPR  (WD / 4)             // 576 float4 chunks per row
#define STAGE_CHUNKS (ROWS * CPR) // 2304 chunks = 8 per thread @288
#define NT   288                  // 9 wave32s; divides both loop trip counts
#define KCH  ((TH * CPR) / NT)    // 4 output chunks per thread per offset

typedef float v4f __attribute__((ext_vector_type(4)));   // native vector: OK for
                                                         // __builtin_nontemporal_store

__global__ __launch_bounds__(NT)
void unfold3d_async_kernel(const float* __restrict__ x, float* __restrict__ out)
{
    __shared__ float tile[ROWS * WD];   // 4 * 2304 * 4 B = 36 KB LDS

    const int tid = threadIdx.x;
    const int nhb = HH / TH;                 // 24 h-slabs
    const int bc  = blockIdx.x / nhb;        // fused (b,c) channel index
    const int hb  = blockIdx.x % nhb;
    const int h0  = hb * TH;

    const float* __restrict__ src = x + (size_t)bc * HWD;

    // ---- Stage ROWS input rows (with zero halo) into LDS.
    // In-range rows: gfx1250 async copy (global_load_async_to_lds_b128, ASYNCcnt).
    // Out-of-range halo rows: zero-fill with ds stores (DScnt).
#pragma unroll
    for (int s = 0; s < STAGE_CHUNKS / NT; ++s) {
        const int idx = tid + s * NT;
        const int r = idx / CPR;             // staged row 0..3
        const int c = idx - r * CPR;         // float4 chunk within row
        const int g = h0 - 1 + r;            // global h of this row
        const unsigned lds_addr = (unsigned)(size_t)(&tile[idx * 4]);
        if ((unsigned)g < (unsigned)HH) {
            const unsigned long long gaddr =
                (unsigned long long)(size_t)(src + (size_t)g * WD + (size_t)c * 4);
            asm volatile("global_load_async_to_lds_b128 %0, %1, off"
                         :: "v"(lds_addr), "v"(gaddr)
                         : "memory");
        } else {
            *(v4f*)(&tile[idx * 4]) = (v4f)(0.0f);
        }
    }

#if __has_builtin(__builtin_amdgcn_s_wait_asynccnt)
    __builtin_amdgcn_s_wait_asynccnt(0);
#else
    asm volatile("s_wait_asynccnt 0" ::: "memory");
#endif
    __syncthreads();   // also drains DScnt for the zero-fill stores

    // ---- Decode this thread's 4 output chunks ONCE (loop-invariant over m).
    int off_[KCH];   // float offset of chunk within the TH-row slab (= LDS base too)
    int w_[KCH];     // w coordinate of chunk
    int d0_[KCH];    // first d of chunk
#pragma unroll
    for (int k = 0; k < KCH; ++k) {
        const int p  = tid + k * NT;
        const int hh = p / CPR;
        const int c4 = p - hh * CPR;
        w_[k]  = c4 / 12;                 // 12 chunks per w-line
        d0_[k] = (c4 - w_[k] * 12) * 4;
        off_[k] = hh * WD + c4 * 4;       // hh*WD + w*48 + d0
    }

    // ---- Emit 27 shifted copies. Full unroll => (i,jj,l) are constants:
    // border checks fold away (all of them for the 3 jj==1,l==1 planes), and the
    // LDS gather base is off_[k] + compile-time shift.
    float* const obc = out + ((size_t)bc * 27) * HWD + (size_t)h0 * WD;

#pragma unroll
    for (int m = 0; m < 27; ++m) {
        const int i  = m / 9;             // h shift component (constant)
        const int jj = (m / 3) % 3;       // w shift component (constant)
        const int l  = m % 3;             // d shift component (constant)
        const int S  = i * WD + (jj - 1) * 48 + (l - 1);   // constant LDS shift
        float* const om = obc + (size_t)m * HWD;

#pragma unroll
        for (int k = 0; k < KCH; ++k) {
            const int  wp   = w_[k] + jj - 1;
            const bool wok  = (unsigned)wp < 48u;          // folds for jj==1
            const int  base = off_[k] + S;

            v4f v;
#pragma unroll
            for (int e = 0; e < 4; ++e) {
                const int  dd = d0_[k] + (l - 1) + e;      // const-folds per (l,e)
                const bool ok = wok & ((unsigned)dd < 48u);
                const int  si = ok ? (base + e) : 0;       // always-safe LDS index
                const float t = tile[si];
                v[e] = ok ? t : 0.0f;
            }
            // 382 MB pure write stream >> 192 MB L2: stream past the caches
            // (global_store_b128 with th:TH_STORE_NT) instead of write-allocating.
            __builtin_nontemporal_store(v, (v4f*)(om + off_[k]));
        }
    }
}

extern "C" void kernel_launch(void* const* d_in, const int* in_sizes, int n_in,
                              void* d_out, int out_size, void* d_ws, size_t ws_size,
                              hipStream_t stream) {
    (void)n_in; (void)out_size; (void)d_ws; (void)ws_size;
    const float* x  = (const float*)d_in[0];
    float* out      = (float*)d_out;

    const int channels = in_sizes[0] / HWD;   // B*C = 32
    const int nblocks  = channels * (HH / TH);

    unfold3d_async_kernel<<<dim3(nblocks), dim3(NT), 0, stream>>>(x, out);
}